// Example1_52055003628298
// MI455X (gfx1250) — compile-verified
//
#include <hip/hip_runtime.h>
#include <hip/hip_bf16.h>

// Problem constants from the reference: b=16, n=1024, v=4096
#define VOCAB   4096
#define SEQ     1024
#define THREADS 256
#define ITEMS   (SEQ / THREADS)   // 4 tokens per thread

// One workgroup per batch row. The whole reference computation collapses to:
//   q = token_ids[b, SEQ-1]
//   scores[m] = R[q, token_ids[b,m]]
//   p = softmax(scores)
//   out[b, :] = scatter_add(p over token_ids[b,:])
__global__ __launch_bounds__(THREADS) void last_row_attn_kernel(
    const int*   __restrict__ token_ids,  // (B, SEQ) int32
    const float* __restrict__ R,          // (VOCAB, VOCAB) f32
    float*       __restrict__ out)        // (B, VOCAB) f32
{
    __shared__ float rowbuf[VOCAB];   // R[q, :]           (16 KB)
    __shared__ float outbuf[VOCAB];   // scatter target    (16 KB)
    __shared__ float red[THREADS];    // block reductions  (1 KB)

    const int b   = blockIdx.x;
    const int tid = threadIdx.x;
    const int* tok = token_ids + b * SEQ;

    // Query token (uniform per block -> scalar load)
    const int q = tok[SEQ - 1];
    const float* rowp = R + (size_t)q * VOCAB;

    // --- CDNA5 async global->LDS copy of the 16 KB row R[q,:] ---
    // GVS mode: the global address is the 64-bit SGPR pair saddr plus the 32-bit byte offset in vaddr,
    //           the LDS destination is the 32-bit byte address held in the vdst VGPR.
    // 256 lanes x 16 B = 4 KB per issue, 4 issues for the full row.
    {
        const unsigned lds_row_base = (unsigned)(size_t)(&rowbuf[0]);
#pragma unroll
        for (int i = 0; i < 4; ++i) {
            unsigned byte_off = (unsigned)tid * 16u + (unsigned)i * 4096u;
            unsigned lds_off  = lds_row_base + byte_off;
            asm volatile("global_load_async_to_lds_b128 %0, %1, %2"
                         :
                         : "v"(lds_off), "v"(byte_off), "s"(rowp)
                         : "memory");
        }
    }

    // Zero the scatter accumulator while the async copy is in flight.
#pragma unroll
    for (int i = 0; i < VOCAB / THREADS; ++i)
        outbuf[tid + i * THREADS] = 0.0f;

    // Fence the async copy (ASYNCcnt), then make LDS visible block-wide.
    asm volatile("s_wait_asynccnt 0" ::: "memory");
    __syncthreads();

    // --- Gather scores from the LDS-resident row ---
    int   t[ITEMS];
    float s[ITEMS];
    float lmax = -3.402823466e38f;
#pragma unroll
    for (int i = 0; i < ITEMS; ++i) {
        t[i] = tok[tid + i * THREADS];      // coalesced VMEM load
        s[i] = rowbuf[t[i]];                // LDS gather
        lmax = fmaxf(lmax, s[i]);
    }

    // --- Block max reduction ---
    red[tid] = lmax;
    __syncthreads();
#pragma unroll
    for (int k = THREADS / 2; k > 0; k >>= 1) {
        if (tid < k) red[tid] = fmaxf(red[tid], red[tid + k]);
        __syncthreads();
    }
    const float gmax = red[0];
    __syncthreads();

    // --- exp + block sum reduction ---
    float e[ITEMS];
    float lsum = 0.0f;
#pragma unroll
    for (int i = 0; i < ITEMS; ++i) {
        e[i] = expf(s[i] - gmax);
        lsum += e[i];
    }
    red[tid] = lsum;
    __syncthreads();
#pragma unroll
    for (int k = THREADS / 2; k > 0; k >>= 1) {
        if (tid < k) red[tid] += red[tid + k];
        __syncthreads();
    }
    const float inv_sum = 1.0f / red[0];
    __syncthreads();

    // --- Scatter softmax weights into LDS (ds_add_f32; tokens may repeat) ---
#pragma unroll
    for (int i = 0; i < ITEMS; ++i)
        atomicAdd(&outbuf[t[i]], e[i] * inv_sum);
    __syncthreads();

    // --- Coalesced store of this batch's logits row ---
    float* outp = out + (size_t)b * VOCAB;
#pragma unroll
    for (int i = 0; i < VOCAB / THREADS; ++i)
        outp[tid + i * THREADS] = outbuf[tid + i * THREADS];
}

extern "C" void kernel_launch(void* const* d_in, const int* in_sizes, int n_in,
                              void* d_out, int out_size, void* d_ws, size_t ws_size,
                              hipStream_t stream) {
    const int*   token_ids = (const int*)d_in[0];   // (B, 1024) int32
    const float* R         = (const float*)d_in[1]; // (4096, 4096) f32
    float*       out       = (float*)d_out;         // (B, 4096) f32

    const int B = in_sizes[0] / SEQ;                // 16

    last_row_attn_kernel<<<B, THREADS, 0, stream>>>(token_ids, R, out);
}